// YOLOLayer_51531017617784
// MI455X (gfx1250) — compile-verified
//
#include <hip/hip_runtime.h>

// YOLO head: conv3x3(256->512) + BN + LeakyReLU -> conv1x1(512->75)+bias -> decode.
// f32 WMMA (V_WMMA_F32_16X16X4_F32) implicit GEMM for both convs, fused epilogues.
// Workspace requirement: 16*512*2704*4 = 88,604,672 bytes in d_ws.

typedef __attribute__((ext_vector_type(2))) float v2f;
typedef __attribute__((ext_vector_type(8))) float v8f;

#define CIN   256
#define CMID  512
#define GRID  52
#define NPIX  2704           // 52*52
#define K1    2304           // 256*9
#define COUT  75
#define STRIDE 8.0f          // 416/52

__device__ __forceinline__ float sigmoid_f(float x) {
    return 1.0f / (1.0f + expf(-x));
}

// ---------------------------------------------------------------------------
// Kernel 1: conv3x3 + BN + LeakyReLU  (implicit GEMM, M=512, N=2704/batch, K=2304)
// block = 256 threads (8 waves). Tile: 64 M x 128 N, 2x2 frags/wave.
// grid = (8, 22, 16)
// ---------------------------------------------------------------------------
#define KC 32                // K chunk staged in LDS
#define NT 128               // N tile
#define LDSTR 36             // padded row stride (floats): even & 16B-friendly

__global__ __launch_bounds__(256)
void conv1_bn_lrelu(const float* __restrict__ xin,
                    const float* __restrict__ w1,
                    const float* __restrict__ gamma,
                    const float* __restrict__ beta,
                    const float* __restrict__ mean,
                    const float* __restrict__ var,
                    float* __restrict__ ws) {
    __shared__ float sA[2][64 * LDSTR];    // A[m][k]  (weights)
    __shared__ float sB[2][NT * LDSTR];    // B[n][k]  (im2col activations)

    const int t      = threadIdx.x;
    const int lane   = t & 31;
    const int l16    = lane & 15;
    const int half   = lane >> 4;
    const int w      = t >> 5;            // wave 0..7
    const int mw     = w & 1;             // 0..1
    const int nw     = w >> 1;            // 0..3

    const int m_base = blockIdx.x * 64;   // output-channel tile
    const int n_base = blockIdx.y * NT;   // spatial tile
    const int bz     = blockIdx.z;        // batch

    const int mA0 = mw * 16;              // m-frag rows (local)
    const int mA1 = mw * 16 + 32;
    const int nB0 = nw * 16;              // n-frag cols (local)
    const int nB1 = nw * 16 + 64;

    v8f acc00 = {};   // (mA0, nB0)
    v8f acc01 = {};   // (mA0, nB1)
    v8f acc10 = {};   // (mA1, nB0)
    v8f acc11 = {};   // (mA1, nB1)

    const float* xb = xin + (size_t)bz * CIN * NPIX;

    auto stage = [&](int buf, int kc) {
        // A: 64 rows x 32 k, contiguous in w1 (row stride K1). 512 float4 loads.
        #pragma unroll
        for (int i = 0; i < 2; ++i) {
            int q  = i * 256 + t;             // 0..511
            int m  = q >> 3;                  // 0..63
            int c4 = (q & 7) << 2;            // 0,4,..,28
            const float4 v4 = *(const float4*)&w1[(size_t)(m_base + m) * K1 + kc + c4];
            *(float4*)&sA[buf][m * LDSTR + c4] = v4;
        }
        // B: gather 32 k x 128 n with halo/bounds. n fastest -> coalesced.
        #pragma unroll
        for (int i = 0; i < 16; ++i) {
            int e = i * 256 + t;              // 0..4095
            int n = e & (NT - 1);
            int k = e >> 7;                   // 0..31
            unsigned kg  = (unsigned)(kc + k);
            unsigned cin = kg / 9u;
            unsigned tap = kg % 9u;
            int ky = (int)(tap / 3u);
            int kx = (int)(tap % 3u);
            int p  = n_base + n;
            float v = 0.0f;
            if (p < NPIX) {
                int y  = (int)((unsigned)p / 52u);
                int xx = (int)((unsigned)p % 52u);
                int iy = y + ky - 1;
                int ix = xx + kx - 1;
                if (iy >= 0 && iy < GRID && ix >= 0 && ix < GRID)
                    v = xb[((size_t)cin * GRID + iy) * GRID + ix];
            }
            sB[buf][n * LDSTR + k] = v;
        }
    };

    stage(0, 0);
    __syncthreads();

    const int NCH = K1 / KC;   // 72
    for (int c = 0; c < NCH; ++c) {
        const int cur = c & 1;
        if (c + 1 < NCH) stage(cur ^ 1, (c + 1) * KC);

        #pragma unroll
        for (int kk = 0; kk < KC; kk += 4) {
            const int ko = kk + 2 * half;
            v2f a0 = *(const v2f*)&sA[cur][(mA0 + l16) * LDSTR + ko];
            v2f a1 = *(const v2f*)&sA[cur][(mA1 + l16) * LDSTR + ko];
            v2f b0 = *(const v2f*)&sB[cur][(nB0 + l16) * LDSTR + ko];
            v2f b1 = *(const v2f*)&sB[cur][(nB1 + l16) * LDSTR + ko];
            acc00 = __builtin_amdgcn_wmma_f32_16x16x4_f32(false, a0, false, b0,
                                                          (short)0, acc00, false, false);
            acc01 = __builtin_amdgcn_wmma_f32_16x16x4_f32(false, a0, false, b1,
                                                          (short)0, acc01, false, false);
            acc10 = __builtin_amdgcn_wmma_f32_16x16x4_f32(false, a1, false, b0,
                                                          (short)0, acc10, false, false);
            acc11 = __builtin_amdgcn_wmma_f32_16x16x4_f32(false, a1, false, b1,
                                                          (short)0, acc11, false, false);
        }
        __syncthreads();
    }

    // Epilogue: BN fold + LeakyReLU(0.1), write [B][CMID][NPIX] f32 workspace
    const int p0 = n_base + nB0 + l16;
    const int p1 = n_base + nB1 + l16;
    #pragma unroll
    for (int part = 0; part < 2; ++part) {
        const int mA = (part == 0) ? mA0 : mA1;
        #pragma unroll
        for (int r = 0; r < 8; ++r) {
            const int o = m_base + mA + r + 8 * half;
            const float scl = gamma[o] * rsqrtf(var[o] + 1e-5f);
            const float sft = beta[o] - mean[o] * scl;
            const size_t row = ((size_t)bz * CMID + o) * NPIX;
            float v0 = ((part == 0) ? acc00[r] : acc10[r]) * scl + sft;
            float v1 = ((part == 0) ? acc01[r] : acc11[r]) * scl + sft;
            v0 = v0 > 0.0f ? v0 : 0.1f * v0;
            v1 = v1 > 0.0f ? v1 : 0.1f * v1;
            if (p0 < NPIX) ws[row + p0] = v0;
            if (p1 < NPIX) ws[row + p1] = v1;
        }
    }
}

// ---------------------------------------------------------------------------
// Kernel 2: conv1x1 (75x512 GEMM) + bias + YOLO decode, fused scatter to output
// block = 256 threads (8 waves). Tile: 80 M x 128 N. grid = (22, 16)
// out layout: [B][A*2704 + p][25]
// ---------------------------------------------------------------------------
__global__ __launch_bounds__(256)
void conv2_yolo(const float* __restrict__ y1,
                const float* __restrict__ w2,
                const float* __restrict__ b2,
                float* __restrict__ out) {
    const int t    = threadIdx.x;
    const int lane = t & 31;
    const int l16  = lane & 15;
    const int half = lane >> 4;
    const int w    = t >> 5;                 // wave 0..7 -> n-group

    const int n_base = blockIdx.x * 128;
    const int bz     = blockIdx.y;

    const int p     = n_base + w * 16 + l16;
    const int p_eff = p < NPIX ? p : (NPIX - 1);

    const float* Bp = y1 + (size_t)bz * CMID * NPIX;

    v8f z = {};
    v8f acc[5] = { z, z, z, z, z };

    #pragma unroll 4
    for (int k0 = 0; k0 < CMID; k0 += 4) {
        const int ko = k0 + 2 * half;
        v2f bf;
        bf.x = Bp[(size_t)ko * NPIX + p_eff];
        bf.y = Bp[(size_t)(ko + 1) * NPIX + p_eff];
        #pragma unroll
        for (int mf = 0; mf < 5; ++mf) {
            const int m = mf * 16 + l16;
            v2f a;
            if (m < COUT) {
                a = *(const v2f*)&w2[(size_t)m * CMID + ko];
            } else {
                a.x = 0.0f; a.y = 0.0f;
            }
            acc[mf] = __builtin_amdgcn_wmma_f32_16x16x4_f32(false, a, false, bf,
                                                            (short)0, acc[mf], false, false);
        }
    }

    if (p < NPIX) {
        const float gx = (float)((unsigned)p % 52u);
        const float gy = (float)((unsigned)p / 52u);
        #pragma unroll
        for (int mf = 0; mf < 5; ++mf) {
            #pragma unroll
            for (int r = 0; r < 8; ++r) {
                const int o = mf * 16 + r + 8 * half;
                if (o < COUT) {
                    const float tv = acc[mf][r] + b2[o];
                    const int a = o / 25;
                    const int c = o - a * 25;
                    const float aw = (a == 0) ? 10.0f : (a == 1) ? 16.0f : 33.0f;
                    const float ah = (a == 0) ? 13.0f : (a == 1) ? 30.0f : 23.0f;
                    float v;
                    if      (c == 0) v = (sigmoid_f(tv) + gx) * STRIDE;
                    else if (c == 1) v = (sigmoid_f(tv) + gy) * STRIDE;
                    else if (c == 2) v = expf(tv) * aw;   // exp(t)*(aw/stride)*stride
                    else if (c == 3) v = expf(tv) * ah;
                    else             v = sigmoid_f(tv);
                    out[((size_t)bz * (3 * NPIX) + (size_t)a * NPIX + p) * 25 + c] = v;
                }
            }
        }
    }
}

extern "C" void kernel_launch(void* const* d_in, const int* in_sizes, int n_in,
                              void* d_out, int out_size, void* d_ws, size_t ws_size,
                              hipStream_t stream) {
    const float* x     = (const float*)d_in[0];
    const float* w1    = (const float*)d_in[1];
    const float* gamma = (const float*)d_in[2];
    const float* beta  = (const float*)d_in[3];
    const float* mn    = (const float*)d_in[4];
    const float* vr    = (const float*)d_in[5];
    const float* w2    = (const float*)d_in[6];
    const float* b2    = (const float*)d_in[7];
    float* ws  = (float*)d_ws;    // needs 16*512*2704*4 bytes
    float* out = (float*)d_out;

    conv1_bn_lrelu<<<dim3(8, 22, 16), 256, 0, stream>>>(x, w1, gamma, beta, mn, vr, ws);
    conv2_yolo<<<dim3(22, 16), 256, 0, stream>>>(ws, w2, b2, out);
}